// SMRNN_59811714564301
// MI455X (gfx1250) — compile-verified
//
#include <hip/hip_runtime.h>
#include <hip/hip_bf16.h>

typedef float v2f __attribute__((ext_vector_type(2)));
typedef float v8f __attribute__((ext_vector_type(8)));

#define IN_N   64
#define OUT_N  64
#define ST_N   256
#define D_N    384   // IN + OUT + ST
#define H_N    1024

// Build inpt = concat(x, prev_output, state) into workspace.
__global__ void smrnn_concat_kernel(const float* __restrict__ x,
                                    const float* __restrict__ po,
                                    const float* __restrict__ st,
                                    float* __restrict__ out) {
    int i = threadIdx.x;
    float v;
    if (i < IN_N)            v = x[i];
    else if (i < IN_N+OUT_N) v = po[i - IN_N];
    else                     v = st[i - IN_N - OUT_N];
    out[i] = v;
}

// y[tile*16 .. tile*16+15] = relu(W[16 rows x K] * x + b)
// One 16-row tile per block; KSPLIT waves split the K dimension and are
// reduced deterministically through LDS. Matrix op: V_WMMA_F32_16X16X4_F32.
//
// fp32 WMMA lane mapping (wave32):
//   A (16x4, M x K): lane m=l&15 is row M, half h=l>>4 selects K pair:
//       VGPR0 = A[m, 2h], VGPR1 = A[m, 2h+1]
//   B (4x16, K x N): lane holds column N=l&15:
//       VGPR0 = B[2h, n], VGPR1 = B[2h+1, n]   (we broadcast x into all n)
//   D (16x16): lane l, VGPR r = D[M = 8h + r, N = l&15]
template <int KSPLIT>
__global__ __launch_bounds__(32 * KSPLIT)
void smrnn_gemv_relu_wmma(const float* __restrict__ W,
                          const float* __restrict__ bias,
                          const float* __restrict__ x,
                          float* __restrict__ y,
                          int K) {
    __shared__ float red[KSPLIT][16];

    const int lane = threadIdx.x & 31;
    const int wave = threadIdx.x >> 5;
    const int tile = blockIdx.x;
    const int m    = lane & 15;   // row within tile (A) / column (B)
    const int h    = lane >> 4;   // lane half -> K sub-pair
    const int row  = tile * 16 + m;

    const int kc   = K / KSPLIT;      // K chunk per wave (multiple of 8)
    const int kbeg = wave * kc;
    const int kend = kbeg + kc;

    const float* __restrict__ wrow = W + (size_t)row * (size_t)K;

    v8f acc0 = {};
    v8f acc1 = {};

    for (int k = kbeg; k < kend; k += 8) {
        // two independent WMMA issues per iteration to hide D->C RAW latency
        v2f a0 = *(const v2f*)(wrow + k + 2 * h);
        v2f b0 = *(const v2f*)(x    + k + 2 * h);
        v2f a1 = *(const v2f*)(wrow + k + 4 + 2 * h);
        v2f b1 = *(const v2f*)(x    + k + 4 + 2 * h);
        __builtin_prefetch(wrow + k + 64, 0, 0);   // global_prefetch_b8, ~8 iters ahead
        acc0 = __builtin_amdgcn_wmma_f32_16x16x4_f32(
                   false, a0, false, b0, (short)0, acc0, false, false);
        acc1 = __builtin_amdgcn_wmma_f32_16x16x4_f32(
                   false, a1, false, b1, (short)0, acc1, false, false);
    }

    v8f c = acc0 + acc1;

    // lanes 0 and 16 of each wave hold the 16 partial outputs of this wave
    if (m == 0) {
        #pragma unroll
        for (int r = 0; r < 8; ++r)
            red[wave][8 * h + r] = c[r];
    }
    __syncthreads();

    if (threadIdx.x < 16) {
        float s = 0.0f;
        #pragma unroll
        for (int w = 0; w < KSPLIT; ++w)
            s += red[w][threadIdx.x];
        s += bias[tile * 16 + threadIdx.x];
        y[tile * 16 + threadIdx.x] = s > 0.0f ? s : 0.0f;
    }
}

extern "C" void kernel_launch(void* const* d_in, const int* in_sizes, int n_in,
                              void* d_out, int out_size, void* d_ws, size_t ws_size,
                              hipStream_t stream) {
    // setup_inputs() order:
    // 0:x 1:w1_0 2:b1_0 3:w1_1 4:b1_1 5:w1_2 6:b1_2
    // 7..16: w2_*/b2_*  17..26: w3_*/b3_*
    // 27:prev_output 28:state 29:prev_weight_change 30:prev_bias_change
    const float* x    = (const float*)d_in[0];
    const float* w1_0 = (const float*)d_in[1];
    const float* b1_0 = (const float*)d_in[2];
    const float* w1_1 = (const float*)d_in[3];
    const float* b1_1 = (const float*)d_in[4];
    const float* w1_2 = (const float*)d_in[5];
    const float* b1_2 = (const float*)d_in[6];
    const float* po   = (const float*)d_in[27];
    const float* st   = (const float*)d_in[28];
    float* out = (float*)d_out;

    float* act0 = (float*)d_ws;          // 384  (inpt)
    float* act1 = act0 + D_N;            // 1024 (layer-0 out)
    float* act2 = act1 + H_N;            // 1024 (layer-1 out)

    smrnn_concat_kernel<<<1, D_N, 0, stream>>>(x, po, st, act0);

    constexpr int KS = 8;                // 8 waves / block, 256 threads
    // layer 0: 384 -> 1024
    smrnn_gemv_relu_wmma<KS><<<H_N / 16, 32 * KS, 0, stream>>>(w1_0, b1_0, act0, act1, D_N);
    // layer 1: 1024 -> 1024
    smrnn_gemv_relu_wmma<KS><<<H_N / 16, 32 * KS, 0, stream>>>(w1_1, b1_1, act1, act2, H_N);
    // layer 2: 1024 -> 64
    smrnn_gemv_relu_wmma<KS><<<OUT_N / 16, 32 * KS, 0, stream>>>(w1_2, b1_2, act2, out, H_N);
}